// Freq2d_43215960933196
// MI455X (gfx1250) — compile-verified
//
#include <hip/hip_runtime.h>

// ---------------------------------------------------------------------------
// Frequency-domain Helmholtz forward modeling (64x64 grid, 32 shots).
//
// A = blocktridiag(cI, T_y, cI), c = 1/h^2 (real),
// T_y = tridiag(c, w2/v^2 - 4c, c) (64x64 complex). Output = u at row ry.
//
// Meet-in-the-middle block elimination (critical path ~32 GJ levels instead
// of ~62; back-substitution steps are cheap WMMA matmuls, so meet at m=32,
// not at ry):
//   up   chain (block 0): D_0=T_0,  D_y = T_y - c^2 D_{y-1}^-1, y = 0..31
//   down chain (block 1): D_63=T_63, D_y = T_y - c^2 D_{y+1}^-1, y = 63..33
//   z chains: z_y = S_y - c G_prev z_prev ; store G_y, z_y history to ws.
// Solve (1 block):
//   (T_32 - c^2 (G_31 + G_33)) u_32 = S_32 - c (G_31 z_31 + G_33 z_33)
//   then back-substitute u_y = G_y (z_y - c u_prev) toward ry (pure WMMA,
//   A-operand streamed from global, next G prefetched).
//
// 64x64 complex inversions: Gauss-Jordan in LDS, 2 barriers/pivot.
// All complex matmuls: V_WMMA_F32_16X16X4_F32, one 16x16 re/im tile pair
// per wave32 (8 waves cover the 64x32 tile).
// ---------------------------------------------------------------------------

typedef __attribute__((ext_vector_type(2))) float v2f;
typedef __attribute__((ext_vector_type(8))) float v8f;

#define GNX  64
#define GNY  64
#define NSH  32
#define MEET 32
#define LDA  66  // padded matrix row stride: bank = (2i+j) mod 64 -> conflict-free

// LDS float offsets (total 12800 floats = 50 KB)
#define MR    0        // 64xLDA matrix real (D or G, in place)
#define MI    4224     // 64xLDA matrix imag
#define ZR    8448     // 64x32 rhs real
#define ZI    10496    // 64x32 rhs imag
#define ROWR  12544    // pivot row snapshot (64)
#define ROWI  12608
#define COLR  12672    // pivot col snapshot (64)
#define COLI  12736
#define LDSF  12800

// ws layout (floats): per-row history, indexed by grid row y (3 MiB total):
//   gws_r = ws            [64 * 4096]
//   gws_i = ws + 262144   [64 * 4096]
//   zws_r = ws + 524288   [64 * 2048]
//   zws_i = ws + 655360   [64 * 2048]

// ---- wave matmul, A from LDS matrix: (accr,acci) += (MR,MI)*(ZR,ZI) -------
__device__ __forceinline__ void cmatmul(const float* lds, int lane, int tm, int tn,
                                        v8f& accr, v8f& acci) {
  const int am = lane & 15;
#pragma unroll
  for (int kk = 0; kk < 64; kk += 4) {
    v2f ar, ai, br, bi, bin;
    const int kb = kk + ((lane >> 4) << 1); // A: K = kb, kb+1
    ar.x = lds[MR + (tm + am) * LDA + kb];
    ar.y = lds[MR + (tm + am) * LDA + kb + 1];
    ai.x = lds[MI + (tm + am) * LDA + kb];
    ai.y = lds[MI + (tm + am) * LDA + kb + 1];
    const int kr = kk + (lane >> 4);        // B: K = kr, kr+2
    br.x = lds[ZR + kr * 32 + tn + am];
    br.y = lds[ZR + (kr + 2) * 32 + tn + am];
    bi.x = lds[ZI + kr * 32 + tn + am];
    bi.y = lds[ZI + (kr + 2) * 32 + tn + am];
    bin.x = -bi.x; bin.y = -bi.y;           // f32 WMMA has no A/B neg
    accr = __builtin_amdgcn_wmma_f32_16x16x4_f32(false, ar, false, br,  (short)0, accr, false, false);
    accr = __builtin_amdgcn_wmma_f32_16x16x4_f32(false, ai, false, bin, (short)0, accr, false, false);
    acci = __builtin_amdgcn_wmma_f32_16x16x4_f32(false, ar, false, bi,  (short)0, acci, false, false);
    acci = __builtin_amdgcn_wmma_f32_16x16x4_f32(false, ai, false, br,  (short)0, acci, false, false);
  }
}

// ---- wave matmul, A streamed from global (stride-64 G planes) -------------
__device__ __forceinline__ void cmatmul_g(const float* __restrict__ gr,
                                          const float* __restrict__ gi,
                                          const float* lds, int lane, int tm, int tn,
                                          v8f& accr, v8f& acci) {
  const int am = lane & 15;
#pragma unroll
  for (int kk = 0; kk < 64; kk += 4) {
    const int kb = kk + ((lane >> 4) << 1);
    const v2f ar = *(const v2f*)(gr + (tm + am) * 64 + kb);   // global_load_b64
    const v2f ai = *(const v2f*)(gi + (tm + am) * 64 + kb);
    v2f br, bi, bin;
    const int kr = kk + (lane >> 4);
    br.x = lds[ZR + kr * 32 + tn + am];
    br.y = lds[ZR + (kr + 2) * 32 + tn + am];
    bi.x = lds[ZI + kr * 32 + tn + am];
    bi.y = lds[ZI + (kr + 2) * 32 + tn + am];
    bin.x = -bi.x; bin.y = -bi.y;
    accr = __builtin_amdgcn_wmma_f32_16x16x4_f32(false, ar, false, br,  (short)0, accr, false, false);
    accr = __builtin_amdgcn_wmma_f32_16x16x4_f32(false, ai, false, bin, (short)0, accr, false, false);
    acci = __builtin_amdgcn_wmma_f32_16x16x4_f32(false, ar, false, bi,  (short)0, acci, false, false);
    acci = __builtin_amdgcn_wmma_f32_16x16x4_f32(false, ai, false, br,  (short)0, acci, false, false);
  }
}

// ---- in-place Gauss-Jordan inversion of 64x64 complex in LDS --------------
__device__ __forceinline__ void gj_invert(float* lds, int tid) {
  const int i  = tid >> 2;
  const int j0 = (tid & 3) << 4;
  for (int k = 0; k < 64; ++k) {
    const float pr  = lds[MR + k * LDA + k];
    const float pi  = lds[MI + k * LDA + k];
    const float idn = 1.0f / (pr * pr + pi * pi);
    const float ipr = pr * idn, ipi = -pi * idn;
    if (tid < 64) {
      lds[COLR + tid] = lds[MR + tid * LDA + k];
      lds[COLI + tid] = lds[MI + tid * LDA + k];
      lds[ROWR + tid] = lds[MR + k * LDA + tid];
      lds[ROWI + tid] = lds[MI + k * LDA + tid];
    }
    __syncthreads();
    if (i == k) {
#pragma unroll
      for (int jj = 0; jj < 16; ++jj) {
        const int j = j0 + jj;
        if (j == k) {
          lds[MR + k * LDA + k] = ipr;
          lds[MI + k * LDA + k] = ipi;
        } else {
          const float xr = lds[ROWR + j], xi = lds[ROWI + j];
          lds[MR + k * LDA + j] = xr * ipr - xi * ipi;
          lds[MI + k * LDA + j] = xr * ipi + xi * ipr;
        }
      }
    } else {
      const float cr = lds[COLR + i], ci = lds[COLI + i];
      const float fr = cr * ipr - ci * ipi;   // f' = col[i] * 1/p
      const float fi = cr * ipi + ci * ipr;
#pragma unroll
      for (int jj = 0; jj < 16; ++jj) {
        const int j = j0 + jj;
        if (j == k) {
          lds[MR + i * LDA + k] = -fr;
          lds[MI + i * LDA + k] = -fi;
        } else {
          const float xr = lds[ROWR + j], xi = lds[ROWI + j];
          lds[MR + i * LDA + j] -= fr * xr - fi * xi;
          lds[MI + i * LDA + j] -= fr * xi + fi * xr;
        }
      }
    }
    __syncthreads();
  }
}

// ---- build D_y = T_y - c2*G_prev in place over MR/MI ----------------------
__device__ __forceinline__ void build_D(float* lds, int tid, const float* vel, int y,
                                        float c, float c2, float w2r, float w2i,
                                        bool first) {
  const int i  = tid >> 2;
  const int j0 = (tid & 3) << 4;
#pragma unroll
  for (int jj = 0; jj < 16; ++jj) {
    const int j = j0 + jj;
    const int e = i * LDA + j;
    float vr = first ? 0.0f : -c2 * lds[MR + e];
    float vi = first ? 0.0f : -c2 * lds[MI + e];
    if (i == j) {
      const float v   = vel[y * GNX + i];
      const float iv2 = 1.0f / (v * v);
      vr += w2r * iv2 - 4.0f * c;
      vi += w2i * iv2;
    } else if (j == i + 1 || j + 1 == i) {
      vr += c;
    }
    lds[MR + e] = vr;
    lds[MI + e] = vi;
  }
}

// ======================= kernel 1: directional chains ======================
__global__ __launch_bounds__(256)
void freq2d_chain(const float* __restrict__ vel, const int* __restrict__ sxs,
                  const int* __restrict__ syp, const int* __restrict__ omegap,
                  const int* __restrict__ ampp, float* __restrict__ ws) {
  __shared__ float lds[LDSF];
  const int tid  = threadIdx.x;
  const int lane = tid & 31;
  const int wv   = tid >> 5;            // wave 0..7 (wave32)
  const int tm   = (wv >> 1) << 4;
  const int tn   = (wv & 1) << 4;
  const int b    = blockIdx.x;          // 0 = up chain, 1 = down chain

  const int   sy  = syp[0];
  const float om  = (float)omegap[0];
  const float amp = (float)ampp[0];
  const float h   = 25.0f;
  const float c   = 1.0f / (h * h);
  const float c2  = c * c;
  const float w2r = om * om * 0.9975f;  // (omega*(1-0.05i))^2
  const float w2i = -om * om * 0.1f;

  float* gws_r = ws;
  float* gws_i = ws + 262144;
  float* zws_r = ws + 524288;
  float* zws_i = ws + 655360;

  const int nsteps = (b == 0) ? MEET : (GNY - 1 - MEET);

  for (int s = 0; s < nsteps; ++s) {
    const int y = (b == 0) ? s : (GNY - 1 - s);
    // z_y = S_y - c * G_prev z_prev  (G_prev = 0 at s==0)
    v8f accr = {}, acci = {};
    if (s > 0) cmatmul(lds, lane, tm, tn, accr, acci);
    __syncthreads();
    {
      const int n    = lane & 15;
      const int moff = (lane >> 4) << 3;
#pragma unroll
      for (int r = 0; r < 8; ++r) {
        const int m  = tm + moff + r;      // grid x
        const int nn = tn + n;             // shot
        float sr = 0.0f;
        if (y == sy && sxs[nn] == m) sr = amp;
        const float zr = sr - c * accr[r];
        const float zi =      -c * acci[r];
        lds[ZR + m * 32 + nn] = zr;
        lds[ZI + m * 32 + nn] = zi;
        zws_r[y * 2048 + m * 32 + nn] = zr;   // z history
        zws_i[y * 2048 + m * 32 + nn] = zi;
      }
    }
    build_D(lds, tid, vel, y, c, c2, w2r, w2i, s == 0);
    __syncthreads();
    gj_invert(lds, tid);                   // M now holds G_y = D_y^-1
    // G history (off the dependency chain: next step only reads M)
    for (int e = tid; e < 4096; e += 256) {
      const int i = e >> 6, j = e & 63;
      gws_r[y * 4096 + e] = lds[MR + i * LDA + j];
      gws_i[y * 4096 + e] = lds[MI + i * LDA + j];
    }
  }
}

// ============ kernel 2: merged solve at row MEET + back-substitution =======
__global__ __launch_bounds__(256)
void freq2d_solve(const float* __restrict__ vel, const int* __restrict__ sxs,
                  const int* __restrict__ syp, const int* __restrict__ ryp,
                  const int* __restrict__ omegap, const int* __restrict__ ampp,
                  const float* __restrict__ ws, float* __restrict__ out) {
  __shared__ float lds[LDSF];
  const int tid  = threadIdx.x;
  const int lane = tid & 31;
  const int wv   = tid >> 5;
  const int tm   = (wv >> 1) << 4;
  const int tn   = (wv & 1) << 4;

  const int   sy  = syp[0];
  const int   ry  = ryp[0];
  const float om  = (float)omegap[0];
  const float amp = (float)ampp[0];
  const float h   = 25.0f;
  const float c   = 1.0f / (h * h);
  const float c2  = c * c;
  const float w2r = om * om * 0.9975f;
  const float w2i = -om * om * 0.1f;

  const float* gws_r = ws;
  const float* gws_i = ws + 262144;
  const float* zws_r = ws + 524288;
  const float* zws_i = ws + 655360;

  const int n    = lane & 15;
  const int moff = (lane >> 4) << 3;

  // rhs accumulation: acc = G_31 z_31 + G_33 z_33 (two streamed matmuls)
  v8f accr = {}, acci = {};
  for (int side = 0; side < 2; ++side) {
    const int ym = (side == 0) ? (MEET - 1) : (MEET + 1);
    for (int e = tid; e < 2048; e += 256) {
      lds[ZR + e] = zws_r[ym * 2048 + e];
      lds[ZI + e] = zws_i[ym * 2048 + e];
    }
    __syncthreads();
    cmatmul_g(gws_r + ym * 4096, gws_i + ym * 4096, lds, lane, tm, tn, accr, acci);
    __syncthreads();
  }

  // rhs = S_m - c*acc ;  M = T_m - c^2 (G_31 + G_33)
  {
#pragma unroll
    for (int r = 0; r < 8; ++r) {
      const int m  = tm + moff + r;
      const int nn = tn + n;
      float sr = 0.0f;
      if (MEET == sy && sxs[nn] == m) sr = amp;
      lds[ZR + m * 32 + nn] = sr - c * accr[r];
      lds[ZI + m * 32 + nn] =      -c * acci[r];
    }
    const int i  = tid >> 2;
    const int j0 = (tid & 3) << 4;
    const float* gur = gws_r + (MEET - 1) * 4096;
    const float* gui = gws_i + (MEET - 1) * 4096;
    const float* gdr = gws_r + (MEET + 1) * 4096;
    const float* gdi = gws_i + (MEET + 1) * 4096;
#pragma unroll
    for (int jj = 0; jj < 16; ++jj) {
      const int j = j0 + jj;
      const int e = i * 64 + j;
      float vr = -c2 * (gur[e] + gdr[e]);
      float vi = -c2 * (gui[e] + gdi[e]);
      if (i == j) {
        const float v   = vel[MEET * GNX + i];
        const float iv2 = 1.0f / (v * v);
        vr += w2r * iv2 - 4.0f * c;
        vi += w2i * iv2;
      } else if (j == i + 1 || j + 1 == i) {
        vr += c;
      }
      lds[MR + i * LDA + j] = vr;
      lds[MI + i * LDA + j] = vi;
    }
  }
  __syncthreads();

  gj_invert(lds, tid);                 // M = (merged D)^-1

  v8f cur_r = {}, cur_i = {};
  cmatmul(lds, lane, tm, tn, cur_r, cur_i);   // u_MEET
  __syncthreads();

  // back-substitution toward ry: u_y = G_y (z_y - c u_prev), pure WMMA
  const int dir = (ry < MEET) ? -1 : 1;
  const int nb  = (ry < MEET) ? (MEET - ry) : (ry - MEET);
  int y = MEET;
  for (int s2 = 0; s2 < nb; ++s2) {
    y += dir;
    // t = z_y - c * u_prev  (each lane owns its 8 C-layout elements)
#pragma unroll
    for (int r = 0; r < 8; ++r) {
      const int m  = tm + moff + r;
      const int nn = tn + n;
      lds[ZR + m * 32 + nn] = zws_r[y * 2048 + m * 32 + nn] - c * cur_r[r];
      lds[ZI + m * 32 + nn] = zws_i[y * 2048 + m * 32 + nn] - c * cur_i[r];
    }
    // pull the next level's G toward the WGP while this step computes
    if (s2 + 1 < nb) {
      __builtin_prefetch(gws_r + (y + dir) * 4096 + tid * 16, 0, 0);
      __builtin_prefetch(gws_i + (y + dir) * 4096 + tid * 16, 0, 0);
    }
    __syncthreads();
    v8f ur = {}, ui = {};
    cmatmul_g(gws_r + y * 4096, gws_i + y * 4096, lds, lane, tm, tn, ur, ui);
    cur_r = ur; cur_i = ui;
    __syncthreads();
  }

  // output u_ry: frd[shot, x], complex64 interleaved
#pragma unroll
  for (int r = 0; r < 8; ++r) {
    const int m  = tm + moff + r;
    const int nn = tn + n;
    out[(nn * GNX + m) * 2 + 0] = cur_r[r];
    out[(nn * GNX + m) * 2 + 1] = cur_i[r];
  }
}

extern "C" void kernel_launch(void* const* d_in, const int* in_sizes, int n_in,
                              void* d_out, int out_size, void* d_ws, size_t ws_size,
                              hipStream_t stream) {
  const float* vel   = (const float*)d_in[0];
  const int*   sxs   = (const int*)d_in[1];
  const int*   sy    = (const int*)d_in[2];
  const int*   ry    = (const int*)d_in[3];
  const int*   omega = (const int*)d_in[4];
  const int*   amp   = (const int*)d_in[5];
  float*       out   = (float*)d_out;
  float*       ws    = (float*)d_ws;   // needs 3 MiB (G/z history)

  freq2d_chain<<<dim3(2), dim3(256), 0, stream>>>(vel, sxs, sy, omega, amp, ws);
  freq2d_solve<<<dim3(1), dim3(256), 0, stream>>>(vel, sxs, sy, ry, omega, amp, ws, out);
}